// ColorDiversityLoss_48679159333230
// MI455X (gfx1250) — compile-verified
//
#include <hip/hip_runtime.h>
#include <hip/hip_bf16.h>
#include <math.h>

typedef __attribute__((ext_vector_type(2))) float v2f;
typedef __attribute__((ext_vector_type(8))) float v8f;

#define NPIX 9216           // 96*96
#define NB 2                // batch
#define KSEL 8              // NUM_CLUSTERS
#define NTILES (NPIX / 16)  // 576 candidate/query tiles
#define WAVES_PER_BLOCK 8
#define TOTAL_WAVES (NB * NTILES)  // 1152

// ---------------------------------------------------------------------------
// Kernel 1: build the pre-scaled, interleaved WMMA A-feed streams.
//   half 0 feed: { -2*c0[m], -2*c1[m] }   (K0, K1 slots)
//   half 1 feed: { -2*c2[m],  |m|^2   }   (K2, K3 slots)
// One b64 load per lane per tile in the hot loop = the exact v2f A operand.
// ---------------------------------------------------------------------------
__global__ void cdl_prep_kernel(const float* __restrict__ gen,
                                v2f* __restrict__ afeed) {
    int idx = blockIdx.x * blockDim.x + threadIdx.x;
    if (idx < NB * NPIX) {
        int b = idx / NPIX;
        int p = idx - b * NPIX;
        const float* base = gen + (size_t)b * 3 * NPIX;
        float c0 = base[p];
        float c1 = base[NPIX + p];
        float c2 = base[2 * NPIX + p];
        float nr = c0 * c0 + c1 * c1 + c2 * c2;
        v2f f0, f1;
        f0.x = -2.0f * c0; f0.y = -2.0f * c1;
        f1.x = -2.0f * c2; f1.y = nr;
        afeed[(size_t)(b * 2 + 0) * NPIX + p] = f0;
        afeed[(size_t)(b * 2 + 1) * NPIX + p] = f1;
    }
}

// Insert v into ascending-sorted s[0..7], keeping the 8 smallest. Branchless.
__device__ __forceinline__ void insert8(float s[8], float v) {
    float cur = fminf(v, s[7]);  // drop the larger of {v, current max}
    #pragma unroll
    for (int i = 6; i >= 0; --i) {
        float mx = fmaxf(s[i], cur);
        cur = fminf(s[i], cur);
        s[i + 1] = mx;
    }
    s[0] = cur;
}

// process one tile pair: 2 WMMAs + single guarded top-8 update (raw values)
__device__ __forceinline__ void process_pair(const v2f a0, const v2f a1,
                                             const v2f bm, float s[8]) {
    v8f c = {};
    v8f d0 = __builtin_amdgcn_wmma_f32_16x16x4_f32(
        false, a0, false, bm, (short)0, c, false, false);
    v8f d1 = __builtin_amdgcn_wmma_f32_16x16x4_f32(
        false, a1, false, bm, (short)0, c, false, false);
    float m01 = fminf(fminf(d0[0], d0[1]), fminf(d0[2], d0[3]));
    float m23 = fminf(fminf(d0[4], d0[5]), fminf(d0[6], d0[7]));
    float m45 = fminf(fminf(d1[0], d1[1]), fminf(d1[2], d1[3]));
    float m67 = fminf(fminf(d1[4], d1[5]), fminf(d1[6], d1[7]));
    float mn = fminf(fminf(m01, m23), fminf(m45, m67));
    if (mn < s[7]) {  // execz-skipped nearly always
        #pragma unroll
        for (int v = 0; v < 8; ++v) insert8(s, d0[v]);
        #pragma unroll
        for (int v = 0; v < 8; ++v) insert8(s, d1[v]);
    }
}

// ---------------------------------------------------------------------------
// Kernel 2: per-query top-8 via WMMA f32 16x16x4 tiles; selection on RAW wmma
// outputs (D = |m|^2 - 2 x_m.x_n is order-equivalent to distance since the
// per-lane query norm qn is constant); sqrt applied only to the 8 survivors.
// One wave = 16 queries (B-matrix); 576 candidate tiles (A-matrix), unrolled
// x2 with unconditional operand prefetch (last pair peeled).
// A 16x4 f32 layout: lane L holds M = L%16, K = vgpr + 2*(L/16)
// B 4x16  f32 layout: lane L holds N = L%16, K = vgpr + 2*(L/16)
// D 16x16 f32 layout: lane L holds N = L%16, M = vgpr + 8*(L/16)
// ---------------------------------------------------------------------------
__global__ void __launch_bounds__(256)
cdl_topk_kernel(const float* __restrict__ gen,
                const v2f* __restrict__ afeed,
                float* __restrict__ partials) {
    const int tid = threadIdx.x;
    const int lane = tid & 31;
    const int n = lane & 15;   // query slot within tile
    const int h = lane >> 4;   // half-wave id
    const int gwave = blockIdx.x * WAVES_PER_BLOCK + (tid >> 5);
    const int batch = gwave / NTILES;
    const int qtile = gwave - batch * NTILES;

    const float* ch0 = gen + (size_t)batch * 3 * NPIX;
    const float* ch1 = ch0 + NPIX;
    const float* ch2 = ch1 + NPIX;

    // ---- B-matrix (queries), fixed for the whole wave ----
    const int q = qtile * 16 + n;
    const float q0 = ch0[q], q1 = ch1[q], q2 = ch2[q];
    const float qn = q0 * q0 + q1 * q1 + q2 * q2;  // |n|^2, per-lane scalar
    v2f bm;
    bm.x = h ? q2 : q0;        // K2 : K0
    bm.y = h ? 1.0f : q1;      // K3 : K1

    // ---- per-lane A-feed stream (pre-scaled, interleaved) ----
    const v2f* pa = afeed + (size_t)(batch * 2 + h) * NPIX + n;

    // top-8 smallest RAW values (order-equivalent to distances)
    float s[8];
    #pragma unroll
    for (int i = 0; i < 8; ++i) s[i] = 3.0e38f;

    // software pipeline: two tiles in flight, unconditional prefetch
    v2f a0 = pa[0];
    v2f a1 = pa[16];
    for (int ct = 0; ct < NTILES - 2; ct += 2) {
        const v2f c0v = a0, c1v = a1;
        a0 = pa[(ct + 2) * 16];
        a1 = pa[(ct + 3) * 16];
        process_pair(c0v, c1v, bm, s);
    }
    process_pair(a0, a1, bm, s);  // peeled final pair (no prefetch)

    // ---- merge the two half-wave candidate partitions for each query ----
    #pragma unroll
    for (int i = 0; i < 8; ++i) {
        float o = __shfl_xor(s[i], 16, 32);
        if (o < s[7]) insert8(s, o);
    }

    // ---- convert the 8 survivors to true distances and sum ----
    float qsum = 0.0f;
    #pragma unroll
    for (int i = 0; i < 8; ++i)
        qsum += sqrtf(fmaxf(s[i] + qn, 0.0f));
    if (h) qsum = 0.0f;  // halves hold identical merged lists; count once
    #pragma unroll
    for (int off = 16; off >= 1; off >>= 1)
        qsum += __shfl_xor(qsum, off, 32);
    if (lane == 0) partials[gwave] = qsum;
}

// ---------------------------------------------------------------------------
// Kernel 3: deterministic final reduction of the 1152 wave partials
// ---------------------------------------------------------------------------
__global__ void cdl_reduce_kernel(const float* __restrict__ partials,
                                  float* __restrict__ out) {
    __shared__ float sm[256];
    const int tid = threadIdx.x;
    float acc = 0.0f;
    for (int i = tid; i < TOTAL_WAVES; i += 256) acc += partials[i];
    sm[tid] = acc;
    __syncthreads();
    for (int s2 = 128; s2 > 0; s2 >>= 1) {
        if (tid < s2) sm[tid] += sm[tid + s2];
        __syncthreads();
    }
    if (tid == 0)
        out[0] = sm[0] * (-1.0f / (float)(NB * NPIX * KSEL));
}

extern "C" void kernel_launch(void* const* d_in, const int* in_sizes, int n_in,
                              void* d_out, int out_size, void* d_ws, size_t ws_size,
                              hipStream_t stream) {
    (void)in_sizes; (void)n_in; (void)out_size; (void)ws_size;
    const float* gen = (const float*)d_in[0];
    float* out = (float*)d_out;
    v2f* afeed = (v2f*)d_ws;                          // NB*2*NPIX v2f
    float* partials = (float*)(afeed + (size_t)NB * 2 * NPIX);  // TOTAL_WAVES floats

    cdl_prep_kernel<<<(NB * NPIX + 255) / 256, 256, 0, stream>>>(gen, afeed);
    cdl_topk_kernel<<<TOTAL_WAVES / WAVES_PER_BLOCK, 256, 0, stream>>>(gen, afeed, partials);
    cdl_reduce_kernel<<<1, 256, 0, stream>>>(partials, out);
}